// Model_90263032692990
// MI455X (gfx1250) — compile-verified
//
#include <hip/hip_runtime.h>

typedef __attribute__((ext_vector_type(2))) float v2f;
typedef __attribute__((ext_vector_type(8))) float v8f;

__device__ __forceinline__ float sigm_fast(float x) {
    return 1.0f / (1.0f + __expf(-x));
}
__device__ __forceinline__ float tanh_fast(float x) {
    // tanh(x) = 1 - 2/(exp(2x)+1); saturates correctly at +/-1 for large |x|
    return 1.0f - 2.0f / (__expf(2.0f * x) + 1.0f);
}
// Wave-uniform broadcast of lane `srcLane`'s value: lowers to v_readlane_b32
// (no LDS pipe, no dscnt wait) since srcLane is a compile-time constant.
__device__ __forceinline__ float lane_bcast(float v, int srcLane) {
    return __int_as_float(__builtin_amdgcn_readlane(__float_as_int(v), srcLane));
}

// ---------------------------------------------------------------------------
// Phase 1: gates_pre[t][n] = sum_k x[t][k] * W_ih[n][k] + (b_ih[n] + b_hh[n])
// One V_WMMA_F32_16X16X4_F32 per 16 timesteps:
//   A (16x4)  = 16 consecutive rows of x      -> 2 VGPRs/lane
//   B (4x16)  = W_ih^T                        -> 2 VGPRs/lane
//   D (16x16) = gate pre-activations          -> 8 VGPRs/lane
// ---------------------------------------------------------------------------
__global__ void lstm_gates_wmma(const float* __restrict__ x,    // (steps,4) chunk
                                const float* __restrict__ Wih,  // (16,4)
                                const float* __restrict__ bih,  // (16,)
                                const float* __restrict__ bhh,  // (16,)
                                float* __restrict__ gates,      // (chunk,16)
                                int tiles, int steps)
{
    const int wave = (int)((blockIdx.x * blockDim.x + threadIdx.x) >> 5);
    if (wave >= tiles) return;                 // wave-uniform: EXEC stays all-1
    const int lane = (int)(threadIdx.x & 31);
    const int half = lane >> 4;                // 0: K=0..1, 1: K=2..3
    const int l    = lane & 15;                // M for A, N for B/D
    const int k0   = half * 2;

    int row = wave * 16 + l;                   // timestep within chunk
    row = row < steps ? row : (steps - 1);     // clamp tail (no divergence)

    v2f a = *(const v2f*)(x   + (size_t)row * 4 + k0);  // A[m][k0], A[m][k0+1]
    v2f b = *(const v2f*)(Wih + l * 4 + k0);            // B[k][n] = W_ih[n][k]
    v8f c = {};
    c = __builtin_amdgcn_wmma_f32_16x16x4_f32(
            /*neg_a=*/false, a, /*neg_b=*/false, b,
            /*c_mod=*/(short)0, c, /*reuse_a=*/false, /*reuse_b=*/false);

    const float bn = bih[l] + bhh[l];
    const int mbase = wave * 16 + half * 8;
#pragma unroll
    for (int r = 0; r < 8; ++r) {
        gates[(size_t)(mbase + r) * 16 + l] = c[r] + bn;
    }
}

// ---------------------------------------------------------------------------
// Phase 2: serial LSTM scan over one chunk, single wave.
// Lanes 0-15 own gate j = lane&15 (lanes 16-31 mirror them). Gate order:
// i=0..3, f=4..7, g=8..11, o=12..15 (PyTorch order). h/c are replicated as
// scalars in every lane; i/f/g/o gathered via v_readlane (wave-uniform).
// ---------------------------------------------------------------------------
__global__ void lstm_scan(const float* __restrict__ gates,  // (steps,16) pre-act
                          const float* __restrict__ Whh,    // (16,4)
                          const float* __restrict__ fcW,    // (4,4)
                          const float* __restrict__ fcb,    // (4,)
                          float* __restrict__ state,        // 8 f32: h0..3,c0..3
                          float* __restrict__ out,          // 4 f32
                          int steps, int first, int last)
{
    const int lane = (int)(threadIdx.x & 31);
    const int j = lane & 15;
    const float w0 = Whh[j * 4 + 0], w1 = Whh[j * 4 + 1];
    const float w2 = Whh[j * 4 + 2], w3 = Whh[j * 4 + 3];

    float h0, h1, h2, h3, c0, c1, c2, c3;
    if (first) {
        h0 = h1 = h2 = h3 = c0 = c1 = c2 = c3 = 0.0f;
    } else {
        h0 = state[0]; h1 = state[1]; h2 = state[2]; h3 = state[3];
        c0 = state[4]; c1 = state[5]; c2 = state[6]; c3 = state[7];
    }

    const float* gp = gates + j;
    float cur = gp[0];

    auto step = [&](float pre) {
        float g = pre;
        g = fmaf(w0, h0, g); g = fmaf(w1, h1, g);
        g = fmaf(w2, h2, g); g = fmaf(w3, h3, g);
        const float act = (j >= 8 && j < 12) ? tanh_fast(g) : sigm_fast(g);
        // gather i,f,g,o into every lane via v_readlane_b32 (constant lanes)
        const float i0 = lane_bcast(act, 0),  i1 = lane_bcast(act, 1);
        const float i2 = lane_bcast(act, 2),  i3 = lane_bcast(act, 3);
        const float f0 = lane_bcast(act, 4),  f1 = lane_bcast(act, 5);
        const float f2 = lane_bcast(act, 6),  f3 = lane_bcast(act, 7);
        const float g0 = lane_bcast(act, 8),  g1 = lane_bcast(act, 9);
        const float g2 = lane_bcast(act, 10), g3 = lane_bcast(act, 11);
        const float o0 = lane_bcast(act, 12), o1 = lane_bcast(act, 13);
        const float o2 = lane_bcast(act, 14), o3 = lane_bcast(act, 15);
        c0 = fmaf(f0, c0, i0 * g0); c1 = fmaf(f1, c1, i1 * g1);
        c2 = fmaf(f2, c2, i2 * g2); c3 = fmaf(f3, c3, i3 * g3);
        h0 = o0 * tanh_fast(c0); h1 = o1 * tanh_fast(c1);
        h2 = o2 * tanh_fast(c2); h3 = o3 * tanh_fast(c3);
    };

    for (int t = 0; t < steps - 1; ++t) {
        float nxt = gp[16];          // issue next load before computing
        gp += 16;
        // Every 32 steps: 32 lanes prefetch 32 consecutive step-lines (2KB)
        // ~256 steps ahead of the stream into the WGP cache.
        if ((t & 31) == 0 && (t + 256 + 32) < steps) {
            __builtin_prefetch(gates + (size_t)(t + 256 + lane) * 16, 0, 0);
        }
        step(cur);
        cur = nxt;
    }
    step(cur);

    if (lane == 0) {                 // carry (h,c) to next chunk
        state[0] = h0; state[1] = h1; state[2] = h2; state[3] = h3;
        state[4] = c0; state[5] = c1; state[6] = c2; state[7] = c3;
    }

    if (last) {                      // softmax(h) @ fc_W.T + fc_b
        const float m  = fmaxf(fmaxf(h0, h1), fmaxf(h2, h3));
        const float e0 = __expf(h0 - m), e1 = __expf(h1 - m);
        const float e2 = __expf(h2 - m), e3 = __expf(h3 - m);
        const float inv = 1.0f / (e0 + e1 + e2 + e3);
        const float y0 = e0 * inv, y1 = e1 * inv, y2 = e2 * inv, y3 = e3 * inv;
        if (lane < 4) {
            const float* w = fcW + lane * 4;
            out[lane] = fmaf(y0, w[0],
                        fmaf(y1, w[1],
                        fmaf(y2, w[2],
                        fmaf(y3, w[3], fcb[lane]))));
        }
    }
}

extern "C" void kernel_launch(void* const* d_in, const int* in_sizes, int n_in,
                              void* d_out, int out_size, void* d_ws, size_t ws_size,
                              hipStream_t stream) {
    const float* x   = (const float*)d_in[0];   // (T,4)
    const float* Wih = (const float*)d_in[1];   // (16,4)
    const float* Whh = (const float*)d_in[2];   // (16,4)
    const float* bih = (const float*)d_in[3];   // (16,)
    const float* bhh = (const float*)d_in[4];   // (16,)
    const float* fcW = (const float*)d_in[5];   // (4,4)
    const float* fcb = (const float*)d_in[6];   // (4,)
    float* out = (float*)d_out;                 // (4,)
    const int T = in_sizes[0] / 4;

    // Chunk the sequence so the gate buffer fits the workspace.
    long long maxSteps = 16;
    if (ws_size > 4096) {
        maxSteps = (long long)((ws_size - 1024) / (16 * sizeof(float)));
    }
    int chunk = 16;
    while ((long long)chunk * 2 <= maxSteps && chunk * 2 <= T && chunk < (1 << 20)) {
        chunk <<= 1;
    }

    float* gates = (float*)d_ws;
    float* state = gates + (size_t)chunk * 16;  // 8 f32 carry slot

    const int nChunks = (T + chunk - 1) / chunk;
    for (int ci = 0; ci < nChunks; ++ci) {
        const int start = ci * chunk;
        const int steps = (T - start) < chunk ? (T - start) : chunk;
        const int tiles = (steps + 15) / 16;
        const int blocks = (tiles + 7) / 8;     // 8 waves (tiles) per 256-thr block
        lstm_gates_wmma<<<blocks, 256, 0, stream>>>(
            x + (size_t)start * 4, Wih, bih, bhh, gates, tiles, steps);
        lstm_scan<<<1, 32, 0, stream>>>(
            gates, Whh, fcW, fcb, state, out,
            steps, ci == 0 ? 1 : 0, ci == (nChunks - 1) ? 1 : 0);
    }
}